// CurveGraphic2d_2508260901149
// MI455X (gfx1250) — compile-verified
//
#include <hip/hip_runtime.h>
#include <hip/hip_bf16.h>
#include <math.h>

typedef float v2f __attribute__((ext_vector_type(2)));
typedef float v8f __attribute__((ext_vector_type(8)));

#define CANVAS_H 256
#define CANVAS_W 256
#define NUM_SAMPLES 64
#define BATCH 16
#define EPSILONF 1e-6f

// ---------------------------------------------------------------------------
// De Casteljau for cubic Bezier (4 control points), component-wise.
// ---------------------------------------------------------------------------
__device__ inline void bez4(const float* cy, const float* cx, float t,
                            float& oy, float& ox) {
    const float s = 1.0f - t;
    // level 1
    float y0 = s * cy[0] + t * cy[1];
    float y1 = s * cy[1] + t * cy[2];
    float y2 = s * cy[2] + t * cy[3];
    float x0 = s * cx[0] + t * cx[1];
    float x1 = s * cx[1] + t * cx[2];
    float x2 = s * cx[2] + t * cx[3];
    // level 2
    float yy0 = s * y0 + t * y1;
    float yy1 = s * y1 + t * y2;
    float xx0 = s * x0 + t * x1;
    float xx1 = s * x1 + t * x2;
    // level 3
    oy = s * yy0 + t * yy1;
    ox = s * xx0 + t * xx1;
}

// ---------------------------------------------------------------------------
// Stage 1: per-curve arc-length-uniform sampling.
// grid = BATCH blocks, 64 threads each. Writes [B][64] x float4 {sy,sx,|s|^2,1}
// to workspace (augmented WMMA A-operand rows).
// ---------------------------------------------------------------------------
__global__ __launch_bounds__(NUM_SAMPLES)
void curve_sample_kernel(const float* __restrict__ inputs,
                         float* __restrict__ ws) {
    const int b = blockIdx.x;
    const int t = threadIdx.x;          // 0..63

    __shared__ float cy[4], cx[4];
    __shared__ float py_[NUM_SAMPLES], px_[NUM_SAMPLES];
    __shared__ float u[NUM_SAMPLES];

    if (t < 4) {
        cy[t] = inputs[(b * 4 + t) * 2 + 0] * (float)CANVAS_H;
        cx[t] = inputs[(b * 4 + t) * 2 + 1] * (float)CANVAS_W;
    }
    __syncthreads();

    const float t0 = (float)t * (1.0f / 63.0f);
    float sy, sx;
    bez4(cy, cx, t0, sy, sx);
    py_[t] = sy;
    px_[t] = sx;
    __syncthreads();

    // serial cumulative arc length + normalization (tiny: 64 elements)
    if (t == 0) {
        float c = 0.0f;
        u[0] = 0.0f;
        for (int i = 1; i < NUM_SAMPLES; ++i) {
            float dy = py_[i] - py_[i - 1];
            float dx = px_[i] - px_[i - 1];
            c += sqrtf(dy * dy + dx * dx);
            u[i] = c;
        }
        const float inv = 1.0f / (c + EPSILONF);
        for (int i = 0; i < NUM_SAMPLES; ++i) u[i] *= inv;
    }
    __syncthreads();

    // t_arc = interp(t0, xp=u, fp=linspace(0,1,64))
    float ta;
    if (t0 <= u[0]) {
        ta = 0.0f;
    } else if (t0 >= u[NUM_SAMPLES - 1]) {
        ta = 1.0f;
    } else {
        int j = 0;
        for (int i = 1; i < NUM_SAMPLES - 1; ++i)
            if (u[i] <= t0) j = i;          // last knot <= query
        const float x0 = u[j], x1 = u[j + 1];
        const float f0 = (float)j * (1.0f / 63.0f);
        const float f1 = (float)(j + 1) * (1.0f / 63.0f);
        ta = (x1 > x0) ? f0 + (t0 - x0) * (f1 - f0) / (x1 - x0) : f0;
    }

    bez4(cy, cx, ta, sy, sx);

    float4 o;
    o.x = sy;
    o.y = sx;
    o.z = sy * sy + sx * sx;
    o.w = 1.0f;
    reinterpret_cast<float4*>(ws)[b * NUM_SAMPLES + t] = o;
}

// ---------------------------------------------------------------------------
// Stage 2: WMMA distance field + shading.
// One wave32 handles 16 pixels of one batch image:
//   dist^2(s,p) = [sy,sx,|s|^2,1] . [-2py,-2px,1,|p|^2]
// computed as 4x V_WMMA_F32_16X16X4_F32 (16 samples per tile), then
// min-reduced over the 16x16 f32 C/D layout (8 VGPRs + lane-half swap).
// ---------------------------------------------------------------------------
__global__ __launch_bounds__(256)
void curve_render_kernel(const float* __restrict__ ws,
                         const float* __restrict__ widths,
                         const float* __restrict__ aaf,
                         float* __restrict__ out) {
    const int lane = threadIdx.x & 31;
    const int wid  = blockIdx.x * (blockDim.x >> 5) + (threadIdx.x >> 5);
    const int b    = wid >> 12;         // 4096 pixel-tiles per batch
    const int col  = lane & 15;
    const int half = lane >> 4;

    // this wave's 16 pixels: p = tile*16 + col, row-major over canvas
    const int p  = (wid & 4095) * 16 + col;
    const float py = (float)(p >> 8);
    const float px = (float)(p & 255);

    // B operand (pixel column): lanes 0-15 supply K=0,1; lanes 16-31 K=2,3
    v2f bm;
    if (half == 0) {
        bm.x = -2.0f * py;
        bm.y = -2.0f * px;
    } else {
        bm.x = 1.0f;
        bm.y = py * py + px * px;
    }

    // A operand rows come from staged float4 {sy, sx, |s|^2, 1}
    const v2f* wsv = reinterpret_cast<const v2f*>(ws) + (size_t)b * NUM_SAMPLES * 2;

    float m = 3.0e38f;
#pragma unroll
    for (int k = 0; k < 4; ++k) {
        const v2f am = wsv[(k * 16 + col) * 2 + half];
        v8f c = {};
        // D = A x B + 0 : 16x16 tile of squared distances
        c = __builtin_amdgcn_wmma_f32_16x16x4_f32(
                /*neg_a=*/false, am, /*neg_b=*/false, bm,
                /*c_mod=*/(short)0, c, /*reuse_a=*/false, /*reuse_b=*/false);
        // min over the 8 rows held by this lane
        float tm = fminf(fminf(fminf(c[0], c[1]), fminf(c[2], c[3])),
                         fminf(fminf(c[4], c[5]), fminf(c[6], c[7])));
        m = fminf(m, tm);
    }
    // merge rows M and M+8 (lane L <-> L+16 hold the two halves of a column)
    m = fminf(m, __shfl_xor(m, 16, 32));

    const float dist = sqrtf(fmaxf(m, 0.0f));
    float v = dist / widths[b] + EPSILONF;
    v = powf(v, aaf[b]);
    float r = 1.0f - v;
    r = fminf(fmaxf(r, 0.0f), 1.0f);

    // lanes 0..15 carry the result for pixel 'col'; lanes 16..31 are duplicates
    if (half == 0) out[(size_t)wid * 16 + col] = r;
}

// ---------------------------------------------------------------------------
extern "C" void kernel_launch(void* const* d_in, const int* in_sizes, int n_in,
                              void* d_out, int out_size, void* d_ws, size_t ws_size,
                              hipStream_t stream) {
    const float* inputs = (const float*)d_in[0];   // [16,4,2]
    const float* widths = (const float*)d_in[1];   // [16]
    const float* aaf    = (const float*)d_in[2];   // [16]
    float* out = (float*)d_out;                    // [16,256,256]
    float* ws  = (float*)d_ws;                     // needs 16*64*16 = 16 KB

    curve_sample_kernel<<<BATCH, NUM_SAMPLES, 0, stream>>>(inputs, ws);

    // 16 batches * 4096 tiles = 65536 waves; 8 wave32 per 256-thread block
    curve_render_kernel<<<8192, 256, 0, stream>>>(ws, widths, aaf, out);
}